// AttnSum3d_42082089566240
// MI455X (gfx1250) — compile-verified
//
#include <hip/hip_runtime.h>
#include <stdint.h>

// Problem constants from the reference.
#define BB      1024
#define SS      512
#define DD      64
#define THREADS 256   // 8 waves of 32 on CDNA5

typedef __attribute__((ext_vector_type(2))) float v2f;
typedef __attribute__((ext_vector_type(8))) float v8f;

// One workgroup per batch. x[b] (512x64 fp32 = 128 KB) is staged in LDS
// (CDNA5 WGP has 320 KB LDS) via GLOBAL_LOAD_ASYNC_TO_LDS_B128 and read
// three times from there:
//   1) column mean, 2) WMMA matvec attn = X*mean, 3) weighted sum w^T*X.
__global__ __launch_bounds__(THREADS)
void attn_sum3d_kernel(const float* __restrict__ x,
                       const int*   __restrict__ mask,
                       float* __restrict__ out,    // [B, D]
                       float* __restrict__ wout)   // [B, S]
{
    extern __shared__ float smem[];
    float* xs    = smem;              // SS*DD   = 32768 floats (128 KB)
    float* attn  = xs   + SS * DD;    // SS      = 512
    float* wgt   = attn + SS;         // SS      = 512
    float* meanv = wgt  + SS;         // DD      = 64
    float* red   = meanv + DD;        // THREADS = 256

    const int t    = threadIdx.x;
    const int b    = blockIdx.x;
    const int lane = t & 31;
    const int wave = t >> 5;

    // ---- Stage A: async stream x[b] HBM -> LDS, no VGPR staging ----------
    // GV mode: per-lane 64-bit global address VGPR pair, per-lane LDS byte
    // offset VGPR (low 32 bits of the generic LDS pointer = wave-relative
    // LDS address, per the FLAT address truncation rule). Tracked on
    // ASYNCcnt; drained with s_wait_asynccnt before the barrier.
    {
        unsigned long long g =
            (unsigned long long)(uintptr_t)(x + (size_t)b * (SS * DD))
            + (unsigned)t * 16u;
        unsigned lds = (unsigned)(uintptr_t)xs + (unsigned)t * 16u;
        #pragma unroll
        for (int i = 0; i < (SS * DD * 4) / (16 * THREADS); ++i) {  // 32 iters
            asm volatile("global_load_async_to_lds_b128 %0, %1, off"
                         :: "v"(lds), "v"(g)
                         : "memory");
            g   += 16ull * THREADS;   // 4096 B per step
            lds += 16u  * THREADS;
        }
        asm volatile("s_wait_asynccnt 0" ::: "memory");
    }
    __syncthreads();

    // ---- Stage B: column mean over s ------------------------------------
    {
        const int c = t & (DD - 1);
        const int g = t >> 6;                 // 4 row-groups
        float s = 0.0f;
        for (int row = g; row < SS; row += 4)
            s += xs[row * DD + c];
        red[t] = s;
        __syncthreads();
        if (t < DD)
            meanv[t] = (red[t] + red[t + 64] + red[t + 128] + red[t + 192])
                       * (1.0f / (float)SS);
        __syncthreads();
    }

    // ---- Stage C: attn = X * mean via V_WMMA_F32_16X16X4_F32 ------------
    // A: 16x4 fp32 tile of X rows.   B: 4x16 with mean in column N=0 only.
    {
        const int ln   = lane & 15;           // M index within tile
        const int hi   = lane >> 4;           // 0: K={0,1}, 1: K={2,3}
        const int koff = hi * 2;
        #pragma unroll
        for (int tile = 0; tile < 4; ++tile) {
            const int r0   = wave * 64 + tile * 16;
            const int rowm = r0 + ln;
            v8f acc = {};
            #pragma unroll
            for (int k0 = 0; k0 < DD; k0 += 4) {
                v2f a, bm;
                a.x = xs[rowm * DD + k0 + koff + 0];
                a.y = xs[rowm * DD + k0 + koff + 1];
                const float b0 = meanv[k0 + koff + 0];
                const float b1 = meanv[k0 + koff + 1];
                bm.x = (ln == 0) ? b0 : 0.0f;   // B[K=koff  ][N=0]
                bm.y = (ln == 0) ? b1 : 0.0f;   // B[K=koff+1][N=0]
                acc = __builtin_amdgcn_wmma_f32_16x16x4_f32(
                        /*neg_a=*/false, a, /*neg_b=*/false, bm,
                        /*c_mod=*/(short)0, acc,
                        /*reuse_a=*/false, /*reuse_b=*/false);
            }
            // Column N=0 of D lives on lanes 0 (M=r0..r0+7) and 16 (M=r0+8..r0+15).
            if (ln == 0) {
                const int base = r0 + hi * 8;
                #pragma unroll
                for (int v = 0; v < 8; ++v)
                    attn[base + v] = acc[v];
            }
        }
    }
    __syncthreads();

    // ---- Stage D: masked softmax over s ---------------------------------
    const int* mrow = mask + (size_t)b * SS;
    {
        const int   m0 = mrow[t];
        const int   m1 = mrow[t + 256];
        const float v0 = m0 ? attn[t]       : -__builtin_inff();
        const float v1 = m1 ? attn[t + 256] : -__builtin_inff();

        red[t] = fmaxf(v0, v1);
        __syncthreads();
        for (int off = 128; off > 0; off >>= 1) {
            if (t < off) red[t] = fmaxf(red[t], red[t + off]);
            __syncthreads();
        }
        const float mx = red[0];
        __syncthreads();

        const float e0 = m0 ? __expf(v0 - mx) : 0.0f;
        const float e1 = m1 ? __expf(v1 - mx) : 0.0f;
        red[t] = e0 + e1;
        __syncthreads();
        for (int off = 128; off > 0; off >>= 1) {
            if (t < off) red[t] += red[t + off];
            __syncthreads();
        }
        const float sum = red[0];
        __syncthreads();

        const float inv = (sum > 0.0f) ? (1.0f / sum) : 0.0f;
        wgt[t]       = e0 * inv;
        wgt[t + 256] = e1 * inv;
    }
    __syncthreads();

    // emit w output [B,S]
    {
        float* wrow = wout + (size_t)b * SS;
        wrow[t]       = wgt[t];
        wrow[t + 256] = wgt[t + 256];
    }

    // ---- Stage E: out = w^T * X -----------------------------------------
    {
        const int c = t & (DD - 1);
        const int g = t >> 6;
        float s = 0.0f;
        for (int row = g; row < SS; row += 4)
            s += wgt[row] * xs[row * DD + c];
        red[t] = s;
        __syncthreads();
        if (t < DD)
            out[(size_t)b * DD + t] =
                red[t] + red[t + 64] + red[t + 128] + red[t + 192];
    }
}

extern "C" void kernel_launch(void* const* d_in, const int* in_sizes, int n_in,
                              void* d_out, int out_size, void* d_ws, size_t ws_size,
                              hipStream_t stream) {
    const float* x    = (const float*)d_in[0];   // [B,S,D] fp32
    const int*   mask = (const int*)d_in[1];     // [B,S]   int32
    float*       out  = (float*)d_out;           // first B*D floats
    float*       wout = out + (size_t)BB * DD;   // then   B*S floats

    const size_t lds_bytes =
        (size_t)(SS * DD + SS + SS + DD + THREADS) * sizeof(float); // ~136 KB

    attn_sum3d_kernel<<<BB, THREADS, lds_bytes, stream>>>(x, mask, out, wout);
}